// TSP_82523501626067
// MI455X (gfx1250) — compile-verified
//
#include <hip/hip_runtime.h>
#include <hip/hip_bf16.h>
#include <math.h>

typedef __attribute__((ext_vector_type(2))) float v2f;
typedef __attribute__((ext_vector_type(8))) float v8f;

#define B_    16
#define T_    8192
#define D_    256
#define L_    256
#define S_    128
#define SPAN_ 64
#define OUTW  (2 * D_ + 1)   // 513 floats per span row

// workspace layout (floats): [0,256) = w = W1@v ; [256, 256 + B*S*64) = softmax weights
#define WS_W   0
#define WS_WTS 256

// ---------------------------------------------------------------------------
// k0: w[d] = sum_l W1[d,l] * v[l]   (b1 dropped: softmax is shift-invariant)
// ---------------------------------------------------------------------------
__global__ void k0_w1v(const float* __restrict__ W1, const float* __restrict__ v,
                       float* __restrict__ ws) {
    int d = threadIdx.x;            // 256 threads, one per output dim
    float acc = 0.f;
#pragma unroll 4
    for (int l = 0; l < L_; ++l) acc += W1[d * L_ + l] * v[l];
    ws[WS_W + d] = acc;
}

// ---------------------------------------------------------------------------
// k1: per-span alpha + softmax -> weights in workspace.
// One workgroup (8 waves) per span; each wave does 8 tokens, lane covers 8 dims.
// Softmax reductions done with wave32 shuffles (2 barriers total).
// ---------------------------------------------------------------------------
__global__ void k1_alpha_softmax(const float* __restrict__ X,
                                 const float* __restrict__ ws,
                                 float* __restrict__ wts) {
    __shared__ float s_alpha[SPAN_];
    __shared__ float s_red[4];

    int blk  = blockIdx.x;          // 0 .. B*S-1
    int b    = blk >> 7;            // / S_
    int s    = blk & (S_ - 1);
    int tid  = threadIdx.x;
    int lane = tid & 31;
    int wave = tid >> 5;            // 0..7

    // preload this lane's 8-dim slice of w (2 x float4, coalesced)
    const float4* wp = (const float4*)(ws + WS_W) + lane * 2;
    float4 w0 = wp[0], w1 = wp[1];

    size_t rowbase = ((size_t)b * T_ + (size_t)s * SPAN_) * D_;
    for (int j = 0; j < 8; ++j) {
        int tloc = wave * 8 + j;
        const float4* xp = (const float4*)(X + rowbase + (size_t)tloc * D_) + lane * 2;
        float4 x0 = xp[0], x1 = xp[1];
        float p = x0.x * w0.x + x0.y * w0.y + x0.z * w0.z + x0.w * w0.w
                + x1.x * w1.x + x1.y * w1.y + x1.z * w1.z + x1.w * w1.w;
        // wave32 all-reduce
        for (int off = 16; off >= 1; off >>= 1) p += __shfl_xor(p, off, 32);
        if (lane == 0) s_alpha[tloc] = p;
    }
    __syncthreads();

    // waves 0..1 hold the 64 alphas (one per thread)
    float a = (tid < SPAN_) ? s_alpha[tid] : -1e30f;
    if (wave < 2) {
        float m = a;
        for (int off = 16; off >= 1; off >>= 1) m = fmaxf(m, __shfl_xor(m, off, 32));
        if (lane == 0) s_red[wave] = m;
    }
    __syncthreads();
    float mx = fmaxf(s_red[0], s_red[1]);
    float e  = (tid < SPAN_) ? __expf(a - mx) : 0.f;
    if (wave < 2) {
        float sum = e;
        for (int off = 16; off >= 1; off >>= 1) sum += __shfl_xor(sum, off, 32);
        if (lane == 0) s_red[2 + wave] = sum;
    }
    __syncthreads();
    if (tid < SPAN_) {
        float inv = 1.f / (s_red[2] + s_red[3]);
        wts[(size_t)blk * SPAN_ + tid] = e * inv;
    }
}

// ---------------------------------------------------------------------------
// k2: span_rep via V_WMMA_F32_16X16X4_F32.
// WG = (b, group of 16 spans, 64-dim chunk); 4 waves, each owns a 16-dim N-tile.
// D[16 spans x 16 dims] += A[16x4 weights, block-diagonal] * B[4 tokens x 16 dims].
// Outer loop = owning span m0 (16 iters); inner 16 WMMA steps fully unrolled.
// A weights: unconditional LDS broadcast read + register cndmask (no exec games).
// ---------------------------------------------------------------------------
__global__ void k2_pool_wmma(const float* __restrict__ X,
                             const float* __restrict__ wts,
                             float* __restrict__ out) {
    __shared__ float s_w[16 * SPAN_];   // 16 spans x 64 softmax weights = 4 KB

    int blk    = blockIdx.x;
    int nchunk = blk & 3;               // which 64-dim chunk of D
    int sg     = (blk >> 2) & 7;        // which 16-span group
    int b      = blk >> 5;
    int tid    = threadIdx.x;           // 0..127
    int lane   = tid & 31;
    int wave   = tid >> 5;              // 0..3

    size_t wbase = ((size_t)b * S_ + (size_t)sg * 16) * SPAN_;
#pragma unroll
    for (int i = 0; i < 8; ++i) s_w[i * 128 + tid] = wts[wbase + i * 128 + tid];
    __syncthreads();

    int nbase = nchunk * 64 + wave * 16;
    int ncol  = nbase + (lane & 15);
    int khalf = lane >> 4;              // 0: K=k0,k0+1  1: K=k0+2,k0+3
    int mrow  = lane & 15;              // A-matrix row held by this lane
    size_t xbase = (size_t)b * T_ * D_ + (size_t)sg * 1024 * D_;

    v8f acc = {0.f, 0.f, 0.f, 0.f, 0.f, 0.f, 0.f, 0.f};
#pragma unroll 1
    for (int m0 = 0; m0 < 16; ++m0) {
        bool mine = (mrow == m0);                       // hoisted compare
        const float* swm  = s_w + m0 * SPAN_ + khalf * 2;
        const float* xrow = X + xbase + ((size_t)m0 * 64 + khalf * 2) * D_ + ncol;
#pragma unroll
        for (int kl = 0; kl < 64; kl += 4) {
            // A[16x4]: lane = M, VGPR0/1 -> K = khalf*2 + {0,1}
            float wv0 = swm[kl];                         // LDS broadcast read
            float wv1 = swm[kl + 1];
            v2f Av = { mine ? wv0 : 0.f, mine ? wv1 : 0.f };
            // B[4x16]: lane&15 = N, VGPR r / half-wave -> K = khalf*2 + r
            v2f Bv = { xrow[(size_t)kl * D_], xrow[(size_t)kl * D_ + D_] };
            acc = __builtin_amdgcn_wmma_f32_16x16x4_f32(
                      /*neg_a=*/false, Av, /*neg_b=*/false, Bv,
                      /*c_mod=*/(short)0, acc, /*reuse_a=*/false, /*reuse_b=*/false);
        }
    }

    // C/D layout: VGPR r, lane l -> M = r + (l>=16 ? 8 : 0), N = l & 15
    size_t orow0 = (size_t)b * S_ + (size_t)sg * 16;
#pragma unroll
    for (int r = 0; r < 8; ++r) {
        int m = r + khalf * 8;
        out[(orow0 + (size_t)m) * OUTW + D_ + ncol] = acc[r];
    }
}

// ---------------------------------------------------------------------------
// k3: span_ends copy (dims 0..255), phi at col 512, prop_lens tail.
// ---------------------------------------------------------------------------
__global__ void k3_ends_phi_lens(const float* __restrict__ X,
                                 const int* __restrict__ offsets,
                                 float* __restrict__ out) {
    int blk = blockIdx.x;               // 0 .. B*S-1
    int b   = blk >> 7;
    int s   = blk & (S_ - 1);
    int tid = threadIdx.x;              // 0..63

    int start = offsets[((size_t)b * S_ + s) * 2 + 0];
    int end   = offsets[((size_t)b * S_ + s) * 2 + 1];

    size_t src = ((size_t)b * T_ + (size_t)(end - 1)) * D_;
    size_t dst = ((size_t)b * S_ + s) * OUTW;

    float4 xv = *(const float4*)(X + src + tid * 4);   // src is 16B-aligned
    out[dst + tid * 4 + 0] = xv.x;                      // dst row stride 513 -> scalar stores
    out[dst + tid * 4 + 1] = xv.y;
    out[dst + tid * 4 + 2] = xv.z;
    out[dst + tid * 4 + 3] = xv.w;

    if (tid == 0) out[dst + 2 * D_] = (float)(end - start);               // phi
    if (s == 0 && tid == 0) out[(size_t)B_ * S_ * OUTW + b] = (float)S_;  // prop_lens
}

// ---------------------------------------------------------------------------
extern "C" void kernel_launch(void* const* d_in, const int* in_sizes, int n_in,
                              void* d_out, int out_size, void* d_ws, size_t ws_size,
                              hipStream_t stream) {
    (void)in_sizes; (void)n_in; (void)out_size; (void)ws_size;
    const float* X    = (const float*)d_in[0];
    const int*   offs = (const int*)d_in[1];
    const float* W1   = (const float*)d_in[2];
    // d_in[3] = b1: uniform alpha shift, softmax-invariant -> unused.
    const float* v    = (const float*)d_in[4];
    float* out = (float*)d_out;
    float* ws  = (float*)d_ws;
    float* wts = ws + WS_WTS;

    k0_w1v<<<1, 256, 0, stream>>>(W1, v, ws);
    k1_alpha_softmax<<<B_ * S_, 256, 0, stream>>>(X, ws, wts);
    k2_pool_wmma<<<B_ * (S_ / 16) * (D_ / 64), 128, 0, stream>>>(X, wts, out);
    k3_ends_phi_lens<<<B_ * S_, 64, 0, stream>>>(X, offs, out);
}